// Net_71545565217445
// MI455X (gfx1250) — compile-verified
//
#include <hip/hip_runtime.h>

// ---------------------------------------------------------------------------
// RNN text classifier on gfx1250 (MI455X), fp32 end-to-end via
// V_WMMA_F32_16X16X4_F32. Weight (B-operand) slabs are staged into LDS with
// gfx1250 async global->LDS loads (double-buffered, ASYNCcnt), removing the
// 8x redundant per-wave weight fetch and halving VMEM traffic per WMMA.
// ---------------------------------------------------------------------------

typedef __attribute__((ext_vector_type(2))) float v2f;
typedef __attribute__((ext_vector_type(4))) float v4f;
typedef __attribute__((ext_vector_type(8))) float v8f;
typedef __attribute__((ext_vector_type(4))) int   v4i;

#define B_      512
#define T_      256
#define D_      1024
#define E_      300
#define CHUNK   256     // K elements staged per LDS buffer
#define NCHUNK  (D_ / CHUNK)
#define WSTRIDE 260     // LDS row stride (floats): 260 % 64 == 4 -> conflict-free b64
#define ESTRIDE 308     // LDS row stride for 300-wide W_ih: 308 % 64 == 52 -> conflict-free

#if __has_builtin(__builtin_amdgcn_global_load_async_to_lds_b128)
#define USE_ASYNC 1
typedef __attribute__((address_space(1))) v4i* gv4i_ptr;  // global v4i*
typedef __attribute__((address_space(3))) v4i* lv4i_ptr;  // LDS v4i*
#else
#define USE_ASYNC 0
#endif

__device__ __forceinline__ void copy16_to_lds(const float* src, float* lds_dst) {
#if USE_ASYNC
  float* s = const_cast<float*>(src);
  __builtin_amdgcn_global_load_async_to_lds_b128((gv4i_ptr)s, (lv4i_ptr)lds_dst, 0, 0);
#else
  *(v4f*)lds_dst = *(const v4f*)src;   // synchronous fallback
#endif
}

__device__ __forceinline__ void async_wait_all() {
#if USE_ASYNC
#if __has_builtin(__builtin_amdgcn_s_wait_asynccnt)
  __builtin_amdgcn_s_wait_asynccnt(0);
#else
  asm volatile("s_wait_asynccnt 0x0" ::: "memory");
#endif
#endif
}

// One wave computes one 16x16 fp32 tile.
// A (16x4 f32) layout: lanes 0-15 -> K pair {0,1}; lanes 16-31 -> {2,3}. B mirrors.
// C/D layout: VGPR i -> M = 16*tileM + i + 8*(lane>=16), N = lane&15.
template <bool WITH_EMB, bool TANH_ACT>
__global__ __launch_bounds__(256) void gemm_step_kernel(
    const float* __restrict__ A,       // [B_ x D_] activations
    const float* __restrict__ W,       // [D_ x D_] weight (row n = output col n)
    const float* __restrict__ bias_a,  // b_hh (RNN) or b0 (MLP)
    const float* __restrict__ bias_b,  // b_ih (RNN only)
    const int*   __restrict__ x,       // [B_ x T_] token ids (RNN only)
    const float* __restrict__ emb,     // [VOCAB x E_]      (RNN only)
    const float* __restrict__ Wih,     // [D_ x E_]         (RNN only)
    int t,
    float* __restrict__ out)           // [B_ x D_]
{
  __shared__ float sB[2 * 16 * WSTRIDE];  // 33.3 KB double-buffered weight slab
                                          // (reused as 16 x ESTRIDE for W_ih)

  const int tid   = threadIdx.x;
  const int lane  = tid & 31;
  const int wave  = tid >> 5;
  const int l16   = lane & 15;
  const int hi    = lane >> 4;             // 0: K {0,1}, 1: K {2,3}
  const int tileN = blockIdx.x;            // 0..63  (output columns, shared by block)
  const int tileM = blockIdx.y * 8 + wave; // 0..31  (batch rows, per wave)
  const int m  = tileM * 16 + l16;
  const int n0 = tileN * 16;
  const int n  = n0 + l16;

  // Cooperative stage of one 16 x CHUNK weight chunk: 1024 b128 items, 4/thread.
  auto stage_w = [&](int buf, int kc) {
    float* dst0 = sB + buf * 16 * WSTRIDE;
#pragma unroll
    for (int j = 0; j < 4; ++j) {
      const int item = tid + j * 256;
      const int r    = item >> 6;          // 0..15
      const int kb   = (item & 63) << 2;   // 0..252
      copy16_to_lds(W + (n0 + r) * D_ + kc + kb, dst0 + r * WSTRIDE + kb);
    }
  };

  v8f acc = {};
  const float* Arow = A + m * D_ + 2 * hi;

  stage_w(0, 0);
  async_wait_all();
  __syncthreads();

  // --- recurrent / dense part: K = 1024 = 4 chunks, B fed from LDS ---
  for (int c = 0; c < NCHUNK; ++c) {
    if (c + 1 < NCHUNK) stage_w((c + 1) & 1, (c + 1) * CHUNK);
    const float* Bl = sB + (c & 1) * 16 * WSTRIDE + l16 * WSTRIDE + 2 * hi;
    const float* Ak = Arow + c * CHUNK;
#pragma unroll 8
    for (int k = 0; k < CHUNK; k += 4) {
      v2f a = *(const v2f*)(Ak + k);
      v2f b = *(const v2f*)(Bl + k);
      acc = __builtin_amdgcn_wmma_f32_16x16x4_f32(
          false, a, false, b, (short)0, acc, false, false);
    }
    async_wait_all();   // next chunk staged
    __syncthreads();    // safe to overwrite the buffer we just read next iter
  }

  // --- embedding part: K = 300, gathered A rows, W_ih slab from LDS ---
  if (WITH_EMB) {
#pragma unroll
    for (int j = 0; j < 5; ++j) {
      const int item = tid + j * 256;      // 16 rows x 75 b128 = 1200 items
      if (item < 16 * 75) {
        const int r  = item / 75;
        const int kb = (item % 75) << 2;
        copy16_to_lds(Wih + (n0 + r) * E_ + kb, sB + r * ESTRIDE + kb);
      }
    }
    const int tok = x[m * T_ + t];
    const float* Erow = emb + (long)tok * E_ + 2 * hi;
    const float* Bl   = sB + l16 * ESTRIDE + 2 * hi;
    async_wait_all();
    __syncthreads();
#pragma unroll 5
    for (int k = 0; k < E_; k += 4) {
      v2f a = *(const v2f*)(Erow + k);
      v2f b = *(const v2f*)(Bl + k);
      acc = __builtin_amdgcn_wmma_f32_16x16x4_f32(
          false, a, false, b, (short)0, acc, false, false);
    }
  }

  const float bsum = bias_a[n] + (WITH_EMB ? bias_b[n] : 0.0f);
  const int mbase  = tileM * 16 + 8 * hi;
#pragma unroll
  for (int i = 0; i < 8; ++i) {
    float v = acc[i] + bsum;
    v = TANH_ACT ? tanhf(v) : fmaxf(v, 0.0f);
    out[(mbase + i) * D_ + n] = v;
  }
}

// h0 = zeros (workspace is poisoned by the harness).
__global__ __launch_bounds__(256) void zero_kernel(float* __restrict__ p, int n) {
  int i = blockIdx.x * blockDim.x + threadIdx.x;
  if (i < n) p[i] = 0.0f;
}

// Final head: logits = relu(z @ W1^T + b1) (3 classes), then log_softmax.
__global__ __launch_bounds__(256) void head_kernel(
    const float* __restrict__ z,   // [B_ x D_]
    const float* __restrict__ W1,  // [3 x D_]
    const float* __restrict__ b1,  // [3]
    float* __restrict__ out)       // [B_ x 3]
{
  const int lane = threadIdx.x & 31;
  const int wave = threadIdx.x >> 5;
  const int row  = blockIdx.x * 8 + wave;

  const float* zr = z + row * D_;
  float s0 = 0.0f, s1 = 0.0f, s2 = 0.0f;
  for (int k = lane; k < D_; k += 32) {
    const float zv = zr[k];
    s0 += zv * W1[k];
    s1 += zv * W1[D_ + k];
    s2 += zv * W1[2 * D_ + k];
  }
#pragma unroll
  for (int off = 16; off > 0; off >>= 1) {
    s0 += __shfl_down(s0, off, 32);
    s1 += __shfl_down(s1, off, 32);
    s2 += __shfl_down(s2, off, 32);
  }
  if (lane == 0) {
    const float l0 = fmaxf(s0 + b1[0], 0.0f);
    const float l1 = fmaxf(s1 + b1[1], 0.0f);
    const float l2 = fmaxf(s2 + b1[2], 0.0f);
    const float mx  = fmaxf(l0, fmaxf(l1, l2));
    const float lse = mx + logf(expf(l0 - mx) + expf(l1 - mx) + expf(l2 - mx));
    out[row * 3 + 0] = l0 - lse;
    out[row * 3 + 1] = l1 - lse;
    out[row * 3 + 2] = l2 - lse;
  }
}

extern "C" void kernel_launch(void* const* d_in, const int* in_sizes, int n_in,
                              void* d_out, int out_size, void* d_ws, size_t ws_size,
                              hipStream_t stream) {
  (void)in_sizes; (void)n_in; (void)out_size; (void)ws_size;

  const int*   x    = (const int*)  d_in[0];
  // d_in[1] = lengths: unused by the reference
  const float* emb  = (const float*)d_in[2];
  const float* W_ih = (const float*)d_in[3];
  const float* b_ih = (const float*)d_in[4];
  const float* W_hh = (const float*)d_in[5];
  const float* b_hh = (const float*)d_in[6];
  const float* W0   = (const float*)d_in[7];
  const float* b0   = (const float*)d_in[8];
  const float* W1   = (const float*)d_in[9];
  const float* b1   = (const float*)d_in[10];

  float* h0 = (float*)d_ws;            // 512*1024 fp32 = 2 MB
  float* h1 = h0 + B_ * D_;            // 2 MB
  float* zb = h1 + B_ * D_;            // 2 MB (MLP activation)

  zero_kernel<<<(B_ * D_ + 255) / 256, 256, 0, stream>>>(h0, B_ * D_);

  const dim3 grid(D_ / 16, B_ / 16 / 8);  // 64 x 4 blocks, 8 waves each
  const dim3 block(256);

  float* hin  = h0;
  float* hout = h1;
  for (int t = 0; t < T_; ++t) {
    gemm_step_kernel<true, true><<<grid, block, 0, stream>>>(
        hin, W_hh, b_hh, b_ih, x, emb, W_ih, t, hout);
    float* tmp = hin; hin = hout; hout = tmp;
  }
  // T_ = 256 launches -> final hidden state is back in h0 (== hin).

  gemm_step_kernel<false, false><<<grid, block, 0, stream>>>(
      hin, W0, b0, nullptr, nullptr, nullptr, nullptr, 0, zb);

  head_kernel<<<B_ / 8, 256, 0, stream>>>(zb, W1, b1, (float*)d_out);
}